// GPTNeoXAttention_Mask_30459908063615
// MI455X (gfx1250) — compile-verified
//
#include <hip/hip_runtime.h>
#include <hip/hip_bf16.h>
#include <math.h>

// ---------------------------------------------------------------------------
// GPTNeoX attention + H2O mask, fp32 throughout (matches reference dtype).
// Matrix ops use V_WMMA_F32_16X16X4_F32 (wave32 CDNA5 fp32 matrix path).
// Data movement: GLOBAL_LOAD_ASYNC_TO_LDS_B128 double-buffered pipelines
// tracked by ASYNCcnt (s_wait_asynccnt), falling back to sync staging if the
// builtin is absent.
// ---------------------------------------------------------------------------

typedef __attribute__((ext_vector_type(2))) float v2f;
typedef __attribute__((ext_vector_type(4))) float v4f;
typedef __attribute__((ext_vector_type(8))) float v8f;
typedef __attribute__((ext_vector_type(4))) int   v4i;

#define S_LEN 2048
#define HID   4096
#define NH    32
#define HD    128
#define QKV_LD 12288          // 3*HID, row stride of fused qkv buffer
#define ROT_HALF 16           // ROTARY_NDIMS/2
#define HEAVY 204             // int(0.1*2048)
#define RECENT 204

#define NEG_BIG (-3.402823466e38f)

// ---------------- CDNA5 async global->LDS support (probe via __has_builtin) --
#if defined(__has_builtin)
#  if __has_builtin(__builtin_amdgcn_global_load_async_to_lds_b128)
#    define HAVE_ASYNC_LDS 1
#  endif
#endif
#ifndef HAVE_ASYNC_LDS
#  define HAVE_ASYNC_LDS 0
#endif

#if HAVE_ASYNC_LDS
typedef __attribute__((address_space(1))) v4i global_v4i;
typedef __attribute__((address_space(3))) v4i local_v4i;

__device__ __forceinline__ void async_copy16(const float* g, float* l)
{
    __builtin_amdgcn_global_load_async_to_lds_b128(
        (global_v4i*)(g), (local_v4i*)(l), 0, 0);
}
# if __has_builtin(__builtin_amdgcn_s_wait_asynccnt)
#  define ASYNC_WAIT(n) __builtin_amdgcn_s_wait_asynccnt(n)
# else
#  define ASYNC_WAIT(n) asm volatile("s_wait_asynccnt %0" :: "i"(n) : "memory")
# endif
#else
# define ASYNC_WAIT(n) ((void)0)
#endif

// ---------------------------------------------------------------------------
// GEMM: C[m][n] = sum_k A[m][k]*W[n][k] + bias[n]
// A: MxK row-major, W: NxK row-major, C: MxN.
// Block tile 64x128, K-block 32, double-buffered LDS. 8 waves as 2(M)x4(N);
// each wave computes 32x32 via 4 WMMA accumulators (2 A-frags x 2 B-frags):
// 4 wmma per 4 ds_load_b64 per K-step.
// ---------------------------------------------------------------------------
#define BM 64
#define BN 128
#define BK 32
#define GEMM_ASYNC_BATCH 6   // async b128 ops per thread per tile: (512+1024)/256

__device__ __forceinline__ void gemm_load_tile(
    const float* __restrict__ A, const float* __restrict__ W,
    float (*As)[BK], float (*Bs)[BK],
    int m0, int n0, int k0, int K, int tid)
{
#if HAVE_ASYNC_LDS
    #pragma unroll
    for (int i = 0; i < 2; ++i) {                // 512 A-chunks of 16B
        int id = tid + i * 256;
        int r = id >> 3, c = (id & 7) << 2;
        async_copy16(&A[(size_t)(m0 + r) * K + k0 + c], &As[r][c]);
    }
    #pragma unroll
    for (int i = 0; i < 4; ++i) {                // 1024 B-chunks of 16B
        int id = tid + i * 256;
        int r = id >> 3, c = (id & 7) << 2;
        async_copy16(&W[(size_t)(n0 + r) * K + k0 + c], &Bs[r][c]);
    }
#else
    for (int i = tid; i < BM * (BK / 4); i += 256) {
        int r = i >> 3, c = (i & 7) << 2;
        *(v4f*)&As[r][c] = *(const v4f*)&A[(size_t)(m0 + r) * K + k0 + c];
    }
    for (int i = tid; i < BN * (BK / 4); i += 256) {
        int r = i >> 3, c = (i & 7) << 2;
        *(v4f*)&Bs[r][c] = *(const v4f*)&W[(size_t)(n0 + r) * K + k0 + c];
    }
    if (k0 + BK < K) {  // prefetch next K-tile (global_prefetch_b8)
        __builtin_prefetch(&A[(size_t)(m0 + (tid & 63)) * K + k0 + BK], 0, 1);
        __builtin_prefetch(&W[(size_t)(n0 + (tid & 127)) * K + k0 + BK], 0, 1);
    }
#endif
}

__global__ __launch_bounds__(256) void gemm_bias_wmma(
    const float* __restrict__ A, const float* __restrict__ W,
    const float* __restrict__ bias, float* __restrict__ C,
    int M, int N, int K)
{
    __shared__ __align__(16) float As[2][BM][BK];   // 2 x 8 KB
    __shared__ __align__(16) float Bs[2][BN][BK];   // 2 x 16 KB

    const int tid  = threadIdx.x;
    const int lane = tid & 31;
    const int wave = tid >> 5;        // 0..7
    const int wm   = wave >> 2;       // 0..1 : 32-row strip
    const int wn   = wave & 3;        // 0..3 : 32-col strip
    const int lm   = lane & 15;
    const int lk   = (lane >> 4) * 2; // K-pair select

    const int m0 = blockIdx.y * BM;
    const int n0 = blockIdx.x * BN;

    v8f acc00 = {}, acc01 = {}, acc10 = {}, acc11 = {};

    int buf = 0;
    gemm_load_tile(A, W, As[0], Bs[0], m0, n0, 0, K, tid);

    for (int k0 = 0; k0 < K; k0 += BK) {
        const int nxt = k0 + BK;
        if (nxt < K) {
            gemm_load_tile(A, W, As[buf ^ 1], Bs[buf ^ 1], m0, n0, nxt, K, tid);
            ASYNC_WAIT(GEMM_ASYNC_BATCH);   // retire current tile's batch only
        } else {
            ASYNC_WAIT(0);
        }
        __syncthreads();                    // all waves see the staged tile

        #pragma unroll
        for (int ks = 0; ks < BK; ks += 4) {
            v2f a0 = *(const v2f*)&As[buf][wm * 32 + lm][ks + lk];
            v2f a1 = *(const v2f*)&As[buf][wm * 32 + 16 + lm][ks + lk];
            v2f b0 = *(const v2f*)&Bs[buf][wn * 32 + lm][ks + lk];
            v2f b1 = *(const v2f*)&Bs[buf][wn * 32 + 16 + lm][ks + lk];
            acc00 = __builtin_amdgcn_wmma_f32_16x16x4_f32(
                false, a0, false, b0, (short)0, acc00, false, false);
            acc01 = __builtin_amdgcn_wmma_f32_16x16x4_f32(
                false, a0, false, b1, (short)0, acc01, false, false);
            acc10 = __builtin_amdgcn_wmma_f32_16x16x4_f32(
                false, a1, false, b0, (short)0, acc10, false, false);
            acc11 = __builtin_amdgcn_wmma_f32_16x16x4_f32(
                false, a1, false, b1, (short)0, acc11, false, false);
        }
        __syncthreads();                    // done reading before next overwrite
        buf ^= 1;
    }

    // D layout: VGPR j -> row M = j + 8*(lane>=16); col N = lane%16
    const int mbase = m0 + wm * 32 + (lane >> 4) * 8;
    const int nbase = n0 + wn * 32 + lm;
    const float bi0 = bias[nbase];
    const float bi1 = bias[nbase + 16];
    #pragma unroll
    for (int j = 0; j < 8; ++j) {
        C[(size_t)(mbase + j) * N + nbase]           = acc00[j] + bi0;
        C[(size_t)(mbase + j) * N + nbase + 16]      = acc01[j] + bi1;
        C[(size_t)(mbase + 16 + j) * N + nbase]      = acc10[j] + bi0;
        C[(size_t)(mbase + 16 + j) * N + nbase + 16] = acc11[j] + bi1;
    }
}

// ---------------------------------------------------------------------------
// RoPE in place on fused qkv (first 32 dims of q and k per head).
// ---------------------------------------------------------------------------
__global__ void rope_kernel(float* __restrict__ qkv, const int* __restrict__ pos)
{
    const int t = blockIdx.x;
    const int h = threadIdx.x;
    const float tf = (float)pos[t];
    float* q = qkv + (size_t)t * QKV_LD + h * 384;
    float* k = q + HD;
    #pragma unroll
    for (int i = 0; i < ROT_HALF; ++i) {
        const float invf = __expf(-(float)i * 0.5756462732485114210f); // ln(1e4)/16
        const float f = tf * invf;
        const float c = __cosf(f), s = __sinf(f);
        float q1 = q[i], q2 = q[i + ROT_HALF];
        q[i]            = q1 * c - q2 * s;
        q[i + ROT_HALF] = q2 * c + q1 * s;
        float k1 = k[i], k2 = k[i + ROT_HALF];
        k[i]            = k1 * c - k2 * s;
        k[i + ROT_HALF] = k2 * c + k1 * s;
    }
}

// ---------------------------------------------------------------------------
// Attention for one (head, 16-query block). Full 16x2048 score block in LDS
// (CDNA5: 320 KB/WGP) -> exact one-pass softmax. K-tiles are staged per-wave
// via async global->LDS copies (coalesced 512B rows), ping-ponged on
// ASYNCcnt; V is read with coalesced per-lane fragments.
// Block: 128 threads (4 waves). Grid: (128 q-blocks, 32 heads).
// ---------------------------------------------------------------------------
#define QB 16
#define KT_FLOATS (QB * HD)     // 2048 floats per K-tile slab

#if HAVE_ASYNC_LDS
__device__ __forceinline__ void issue_ktile(
    const float* __restrict__ qkv, int h, int kb, float* slab, int lane)
{
    const float* base = qkv + (size_t)(kb * 16) * QKV_LD + h * 384 + HD;
    #pragma unroll
    for (int it = 0; it < 16; ++it)   // one coalesced 512B row per instruction
        async_copy16(base + (size_t)it * QKV_LD + lane * 4, slab + it * HD + lane * 4);
}
#endif

__global__ __launch_bounds__(128) void attn_kernel(
    const float* __restrict__ qkv, const float* __restrict__ amask,
    float* __restrict__ ssum, float* __restrict__ attn_out)
{
    extern __shared__ float lds[];
    float* q_s  = lds;                       // [QB][128]
    float* sc   = q_s + QB * HD;             // [QB][S_LEN]
    float* red  = sc + QB * S_LEN;           // [QB][8]
    float* rowv = red + QB * 8;              // [QB]
    float* kslab = rowv + QB;                // [4 waves][2 bufs][QB*HD], 16B-aligned

    const int h      = blockIdx.y;
    const int qb     = blockIdx.x;
    const int q_base = qb * QB;
    const int tid    = threadIdx.x;
    const int lane   = tid & 31;
    const int wave   = tid >> 5;             // 0..3
    const int lm     = lane & 15;
    const int lkh    = lane >> 4;            // 0/1 half select
    const float inv_norm = 0.08838834764831843876f;  // 1/sqrt(128)

    // ---- stage Q tile in LDS ----
    for (int i = tid; i < QB * HD / 4; i += 128) {
        int r = i >> 5;
        int c = (i & 31) << 2;
        *(v4f*)&q_s[r * HD + c] =
            *(const v4f*)&qkv[(size_t)(q_base + r) * QKV_LD + h * 384 + c];
    }
    __syncthreads();

    // ---- phase 1: scores = Q K^T, waves stride over 16-key blocks ----
#if HAVE_ASYNC_LDS
    float* myslab = kslab + wave * 2 * KT_FLOATS;
    int sl = 0;
    issue_ktile(qkv, h, wave, myslab, lane);
    for (int kb = wave; kb < S_LEN / 16; kb += 4) {
        const int nxt = kb + 4;
        if (nxt < S_LEN / 16) {
            issue_ktile(qkv, h, nxt, myslab + (sl ^ 1) * KT_FLOATS, lane);
            ASYNC_WAIT(16);          // retire current tile's 16-op batch
        } else {
            ASYNC_WAIT(0);
        }
        const float* ks_ = myslab + sl * KT_FLOATS;
        v8f acc = {};
        #pragma unroll
        for (int d = 0; d < HD; d += 4) {
            v2f a = *(const v2f*)&q_s[lm * HD + d + 2 * lkh];
            v2f b = *(const v2f*)&ks_[lm * HD + d + 2 * lkh];
            acc = __builtin_amdgcn_wmma_f32_16x16x4_f32(
                false, a, false, b, (short)0, acc, false, false);
        }
        const int key = kb * 16 + lm;
        const float am = amask[key];
        #pragma unroll
        for (int j = 0; j < 8; ++j) {
            const int qrow = j + 8 * lkh;
            sc[qrow * S_LEN + key] =
                (key <= q_base + qrow) ? (acc[j] * inv_norm + am) : NEG_BIG;
        }
        sl ^= 1;
    }
#else
    for (int kb = wave; kb < S_LEN / 16; kb += 4) {
        v8f acc = {};
        const float* kbase = qkv + (size_t)(kb * 16 + lm) * QKV_LD + h * 384 + HD;
        #pragma unroll
        for (int d = 0; d < HD; d += 4) {
            v2f a = *(const v2f*)&q_s[lm * HD + d + 2 * lkh];
            v2f b = *(const v2f*)&kbase[d + 2 * lkh];
            acc = __builtin_amdgcn_wmma_f32_16x16x4_f32(
                false, a, false, b, (short)0, acc, false, false);
        }
        const int key = kb * 16 + lm;
        const float am = amask[key];
        #pragma unroll
        for (int j = 0; j < 8; ++j) {
            const int qrow = j + 8 * lkh;
            sc[qrow * S_LEN + key] =
                (key <= q_base + qrow) ? (acc[j] * inv_norm + am) : NEG_BIG;
        }
    }
#endif
    __syncthreads();

    // ---- phase 2: softmax; 8 threads per query row ----
    const int r  = tid >> 3;
    const int l8 = tid & 7;
    float mx = NEG_BIG;
    for (int k2 = l8; k2 < S_LEN; k2 += 8) mx = fmaxf(mx, sc[r * S_LEN + k2]);
    red[r * 8 + l8] = mx;
    __syncthreads();
    if (l8 == 0) {
        float m2 = red[r * 8];
        #pragma unroll
        for (int i = 1; i < 8; ++i) m2 = fmaxf(m2, red[r * 8 + i]);
        rowv[r] = m2;
    }
    __syncthreads();
    const float rmax = rowv[r];
    float sum = 0.f;
    for (int k2 = l8; k2 < S_LEN; k2 += 8) {
        float e = __expf(sc[r * S_LEN + k2] - rmax);
        sc[r * S_LEN + k2] = e;
        sum += e;
    }
    red[r * 8 + l8] = sum;
    __syncthreads();
    if (l8 == 0) {
        float s2 = 0.f;
        #pragma unroll
        for (int i = 0; i < 8; ++i) s2 += red[r * 8 + i];
        rowv[r] = s2;
    }
    __syncthreads();
    const float rinv = 1.0f / rowv[r];
    for (int k2 = l8; k2 < S_LEN; k2 += 8) sc[r * S_LEN + k2] *= rinv;
    __syncthreads();

    // ---- phase 3: scores_sum[h,k] += sum_q weights[q,k] ----
    for (int k2 = tid; k2 < S_LEN; k2 += 128) {
        float s2 = 0.f;
        #pragma unroll
        for (int qq = 0; qq < QB; ++qq) s2 += sc[qq * S_LEN + k2];
        atomicAdd(&ssum[h * S_LEN + k2], s2);
    }

    // ---- phase 4: O = weights @ V ; each wave owns two 16-dim tiles ----
    #pragma unroll
    for (int nt = 0; nt < 2; ++nt) {
        const int d0 = (wave * 2 + nt) * 16;
        const float* vbase = qkv + (size_t)(h * 384 + 256 + d0 + lm);
        v8f acc = {};
        for (int kb = 0; kb < S_LEN; kb += 4) {
            v2f a = *(const v2f*)&sc[lm * S_LEN + kb + 2 * lkh];
            const int t0 = kb + 2 * lkh;
            v2f b;
            b.x = vbase[(size_t)t0 * QKV_LD];
            b.y = vbase[(size_t)(t0 + 1) * QKV_LD];
            acc = __builtin_amdgcn_wmma_f32_16x16x4_f32(
                false, a, false, b, (short)0, acc, false, false);
        }
        #pragma unroll
        for (int j = 0; j < 8; ++j) {
            const int t = q_base + j + 8 * lkh;
            attn_out[(size_t)t * HID + h * HD + d0 + lm] = acc[j];
        }
    }
}

// ---------------------------------------------------------------------------
// H2O mask: base recent-window mask + top-204 heavy hitters over the first
// 1844 columns of scores_sum. One block (256 threads) per head.
// ---------------------------------------------------------------------------
__global__ __launch_bounds__(256) void hh_mask_kernel(
    const float* __restrict__ ssum, float* __restrict__ mask_out)
{
    __shared__ float vals[S_LEN];
    __shared__ float rv[256];
    __shared__ int   ri[256];

    const int h   = blockIdx.x;
    const int tid = threadIdx.x;
    const int SEL = S_LEN - RECENT;   // 1844 candidate columns

    for (int c = tid; c < S_LEN + 1; c += 256)
        mask_out[h * (S_LEN + 1) + c] = (c >= S_LEN + 1 - RECENT) ? 1.0f : 0.0f;
    for (int i = tid; i < S_LEN; i += 256)
        vals[i] = (i < SEL) ? ssum[h * S_LEN + i] : NEG_BIG;
    __syncthreads();

    for (int it = 0; it < HEAVY; ++it) {
        float bv = NEG_BIG;
        int   bi = 0;
        for (int i = tid; i < SEL; i += 256) {
            float v = vals[i];
            if (v > bv) { bv = v; bi = i; }
        }
        rv[tid] = bv; ri[tid] = bi;
        __syncthreads();
        for (int s2 = 128; s2 > 0; s2 >>= 1) {
            if (tid < s2 && rv[tid + s2] > rv[tid]) {
                rv[tid] = rv[tid + s2]; ri[tid] = ri[tid + s2];
            }
            __syncthreads();
        }
        if (tid == 0) {
            vals[ri[0]] = NEG_BIG;
            mask_out[h * (S_LEN + 1) + ri[0]] = 1.0f;
        }
        __syncthreads();
    }
}

__global__ void zero_kernel(float* __restrict__ p, int n)
{
    int i = blockIdx.x * 256 + threadIdx.x;
    if (i < n) p[i] = 0.f;
}

// ---------------------------------------------------------------------------
extern "C" void kernel_launch(void* const* d_in, const int* in_sizes, int n_in,
                              void* d_out, int out_size, void* d_ws, size_t ws_size,
                              hipStream_t stream)
{
    const float* hidden  = (const float*)d_in[0];   // [1,2048,4096]
    const float* amask   = (const float*)d_in[1];   // [1,1,1,2048]
    const int*   pos     = (const int*)d_in[2];     // [1,2048]
    const float* qkv_w   = (const float*)d_in[3];   // [12288,4096]
    const float* qkv_b   = (const float*)d_in[4];   // [12288]
    const float* dense_w = (const float*)d_in[5];   // [4096,4096]
    const float* dense_b = (const float*)d_in[6];   // [4096]

    float* out_attn = (float*)d_out;                          // 2048*4096
    float* out_mask = (float*)d_out + (size_t)S_LEN * HID;    // 32*2049

    float* qkv     = (float*)d_ws;                            // 2048*12288
    float* attn_ws = qkv + (size_t)S_LEN * QKV_LD;            // 2048*4096
    float* ssum    = attn_ws + (size_t)S_LEN * HID;           // 32*2048

    zero_kernel<<<(NH * S_LEN + 255) / 256, 256, 0, stream>>>(ssum, NH * S_LEN);

    // QKV projection: [2048,4096] x [12288,4096]^T + b
    gemm_bias_wmma<<<dim3(QKV_LD / BN, S_LEN / BM), 256, 0, stream>>>(
        hidden, qkv_w, qkv_b, qkv, S_LEN, QKV_LD, HID);

    rope_kernel<<<S_LEN, NH, 0, stream>>>(qkv, pos);

    size_t lds_bytes =
        (size_t)(QB * HD + QB * S_LEN + QB * 8 + QB) * sizeof(float);
#if HAVE_ASYNC_LDS
    lds_bytes += (size_t)4 * 2 * KT_FLOATS * sizeof(float);   // per-wave K slabs
#endif
    attn_kernel<<<dim3(S_LEN / QB, NH), 128, lds_bytes, stream>>>(
        qkv, amask, ssum, attn_ws);

    hh_mask_kernel<<<NH, 256, 0, stream>>>(ssum, out_mask);

    // Dense projection: [2048,4096] x [4096,4096]^T + b
    gemm_bias_wmma<<<dim3(HID / BN, S_LEN / BM), 256, 0, stream>>>(
        attn_ws, dense_w, dense_b, out_attn, S_LEN, HID, HID);
}